// SLA_57784490000880
// MI455X (gfx1250) — compile-verified
//
#include <hip/hip_runtime.h>
#include <hip/hip_bf16.h>

// SLA forward for MI455X (gfx1250, wave32).
// All contractions run on the CDNA5 matrix pipe via V_WMMA_F32_16X16X4_F32
// (fp32 A/B, fp32 accum) to stay numerically faithful to the fp32 reference
// while exercising v_wmma. Problem is ~22 GFLOP / ~300MB traffic -> matrix
// pipe bound on a 23.3 TB/s part; fp32 WMMA throughput is ample.

typedef __attribute__((ext_vector_type(2))) float v2f;
typedef __attribute__((ext_vector_type(8))) float v8f;

#define B_   8
#define C_   256
#define L_   4096
#define NH_  4
#define D_   64
#define NQ_  64
#define NK_  64
#define T_   6

__device__ __forceinline__ v8f wmma4(v2f a, v2f b, v8f c) {
  // D = A(16x4 f32) * B(4x16 f32) + C(16x16 f32)
  return __builtin_amdgcn_wmma_f32_16x16x4_f32(false, a, false, b,
                                               (short)0, c, false, false);
}
__device__ __forceinline__ v8f vzero8() {
  v8f z; for (int i = 0; i < 8; ++i) z[i] = 0.f; return z;
}

// ---------------------------------------------------------------------------
// K1: qkv = qkv_w(768x256) @ x(b, 256, 4096); writes q/k/v in (B,NH,L,D).
// Tile: 64(M=o) x 64(N=l), K=256 in 4 LDS-staged chunks of 64.
// ---------------------------------------------------------------------------
__global__ __launch_bounds__(256) void k_qkv(
    const float* __restrict__ x, const float* __restrict__ w,
    float* __restrict__ qO, float* __restrict__ kO, float* __restrict__ vO) {
  const int b = blockIdx.z, oy = blockIdx.y, lx = blockIdx.x;
  const int o0 = oy * 64, l0 = lx * 64;
  __shared__ float lW[64 * 65];
  __shared__ float lX[64 * 65];
  const int t = threadIdx.x, lane = t & 31, wave = t >> 5;
  const int fm = lane & 15, fkb = (lane >> 4) << 1;
  v8f acc[2]; acc[0] = vzero8(); acc[1] = vzero8();

  for (int kc = 0; kc < 4; ++kc) {
    __syncthreads();
    for (int r = 0; r < 16; ++r) {
      int idx = r * 256 + t;
      int i = idx >> 6, j = idx & 63;
      lW[i * 65 + j] = w[(size_t)(o0 + i) * C_ + kc * 64 + j];
      lX[i * 65 + j] = x[((size_t)b * C_ + kc * 64 + i) * L_ + l0 + j];
      if (kc < 3)  // prefetch next K-chunk of X (global_prefetch_b8)
        __builtin_prefetch(&x[((size_t)b * C_ + (kc + 1) * 64 + i) * L_ + l0 + j], 0, 0);
    }
    __syncthreads();
    for (int kk = 0; kk < 64; kk += 4) {
      for (int s = 0; s < 2; ++s) {
        int sub = wave * 2 + s;
        int m0 = (sub >> 2) << 4, n0 = (sub & 3) << 4;
        v2f a, bb;
        a.x = lW[(m0 + fm) * 65 + kk + fkb];
        a.y = lW[(m0 + fm) * 65 + kk + fkb + 1];
        bb.x = lX[(kk + fkb) * 65 + n0 + fm];
        bb.y = lX[(kk + fkb + 1) * 65 + n0 + fm];
        acc[s] = wmma4(a, bb, acc[s]);
      }
    }
  }
  const int part = oy >> 2, h = oy & 3;
  float* dst = (part == 0) ? qO : ((part == 1) ? kO : vO);
  dst += ((size_t)b * NH_ + h) * (size_t)L_ * D_;
  const int N = lane & 15, mofs = (lane < 16) ? 0 : 8;
  for (int s = 0; s < 2; ++s) {
    int sub = wave * 2 + s;
    int m0 = (sub >> 2) << 4, n0 = (sub & 3) << 4;
    size_t l = (size_t)(l0 + n0 + N);
    for (int r = 0; r < 8; ++r)
      dst[l * D_ + (m0 + mofs + r)] = acc[s][r];
  }
}

// ---------------------------------------------------------------------------
// K2: block means qb/kb: (B,NH,nQ,D) = mean over 64 rows.
// ---------------------------------------------------------------------------
__global__ __launch_bounds__(64) void k_bmean(
    const float* __restrict__ q, const float* __restrict__ k,
    float* __restrict__ qb, float* __restrict__ kb) {
  const int g = blockIdx.x, d = threadIdx.x;        // g = bh*64 + n
  const float* qp = q + (size_t)g * 4096;
  const float* kp = k + (size_t)g * 4096;
  float sq = 0.f, sk = 0.f;
  for (int r = 0; r < 64; ++r) { sq += qp[r * 64 + d]; sk += kp[r * 64 + d]; }
  qb[(size_t)g * 64 + d] = sq * (1.f / 64.f);
  kb[(size_t)g * 64 + d] = sk * (1.f / 64.f);
}

// ---------------------------------------------------------------------------
// K3: blk_scores = qb @ kb^T per (b,h); top-6 per row (strict > keeps the
// lowest index first on ties, matching jax.lax.top_k).
// ---------------------------------------------------------------------------
__global__ __launch_bounds__(64) void k_topk(
    const float* __restrict__ qb, const float* __restrict__ kb,
    int* __restrict__ lut) {
  const int bh = blockIdx.x, t = threadIdx.x;
  __shared__ float sq[64 * 65];
  __shared__ float sk[64 * 65];
  for (int r = 0; r < 64; ++r) {
    sq[r * 65 + t] = qb[(size_t)bh * 4096 + r * 64 + t];
    sk[r * 65 + t] = kb[(size_t)bh * 4096 + r * 64 + t];
  }
  __syncthreads();
  float best[T_]; int bidx[T_];
  for (int i = 0; i < T_; ++i) { best[i] = -3.0e38f; bidx[i] = 0; }
  for (int m = 0; m < 64; ++m) {
    float s = 0.f;
    for (int d = 0; d < 64; ++d) s += sq[t * 65 + d] * sk[m * 65 + d];
    if (s > best[T_ - 1]) {
      int p = T_ - 1;
      while (p > 0 && s > best[p - 1]) {
        best[p] = best[p - 1]; bidx[p] = bidx[p - 1]; --p;
      }
      best[p] = s; bidx[p] = m;
    }
  }
  for (int i = 0; i < T_; ++i) lut[((size_t)bh * 64 + t) * T_ + i] = bidx[i];
}

// ---------------------------------------------------------------------------
// K4: sparse attention. One WG per (b,h,n,half): 32 q-rows x 384 gathered
// keys. WMMA QK^T -> LDS softmax -> WMMA P*V. Writes o_s.
// ---------------------------------------------------------------------------
__global__ __launch_bounds__(256) void k_sattn(
    const float* __restrict__ q, const float* __restrict__ k,
    const float* __restrict__ v, const int* __restrict__ lut,
    float* __restrict__ o) {
  const int g = blockIdx.x;
  const int half = g & 1, bhn = g >> 1;
  const int n = bhn & 63, bh = bhn >> 6;
  const int qrow0 = n * 64 + half * 32;
  const float* qp = q + (size_t)bh * L_ * D_;
  const float* kp = k + (size_t)bh * L_ * D_;
  const float* vp = v + (size_t)bh * L_ * D_;
  __shared__ float sS[32 * 385];   // scores/probs, padded
  __shared__ float sQ[32 * 65];
  __shared__ int   sLut[T_];
  const int t = threadIdx.x, lane = t & 31, wave = t >> 5;
  const int fm = lane & 15, fkb = (lane >> 4) << 1;
  if (t < T_) sLut[t] = lut[(size_t)bhn * T_ + t];
  for (int r = 0; r < 8; ++r) {
    int idx = r * 256 + t;
    int i = idx >> 6, j = idx & 63;
    sQ[i * 65 + j] = qp[(size_t)(qrow0 + i) * D_ + j];
  }
  __syncthreads();
  // Phase 1: S = Q * Kg^T  (32 x 384) -> 48 subtiles, 6 per wave.
  for (int si = wave * 6; si < wave * 6 + 6; ++si) {
    const int m0 = (si / 24) * 16;
    const int kg0 = (si % 24) * 16;
    const int krow = sLut[kg0 >> 6] * 64 + (kg0 & 63) + fm;  // gathered key row
    v8f acc = vzero8();
    for (int kk = 0; kk < 64; kk += 4) {
      v2f a, bb;
      a.x = sQ[(m0 + fm) * 65 + kk + fkb];
      a.y = sQ[(m0 + fm) * 65 + kk + fkb + 1];
      const float* kr = kp + (size_t)krow * D_ + kk + fkb;
      bb.x = kr[0]; bb.y = kr[1];
      acc = wmma4(a, bb, acc);
    }
    const int N = lane & 15, mofs = (lane < 16) ? 0 : 8;
    for (int r = 0; r < 8; ++r) sS[(m0 + mofs + r) * 385 + kg0 + N] = acc[r];
  }
  __syncthreads();
  // Phase 2: row softmax over 384 with scale D^-1/2 = 1/8.
  if (t < 32) {
    const float scale = 0.125f;
    float mx = -3.0e38f;
    for (int j = 0; j < 384; ++j) {
      float vv = sS[t * 385 + j] * scale; sS[t * 385 + j] = vv;
      if (vv > mx) mx = vv;
    }
    float sum = 0.f;
    for (int j = 0; j < 384; ++j) {
      float e = __expf(sS[t * 385 + j] - mx); sS[t * 385 + j] = e; sum += e;
    }
    float inv = 1.f / sum;
    for (int j = 0; j < 384; ++j) sS[t * 385 + j] *= inv;
  }
  __syncthreads();
  // Phase 3: O = P * Vg (32 x 64), 8 subtiles, 1 per wave, K=384.
  {
    const int m0 = (wave >> 2) * 16, n0 = (wave & 3) * 16;
    v8f acc = vzero8();
    for (int kk = 0; kk < 384; kk += 4) {
      const int kg = kk + fkb;                  // kk..kk+3 within one block
      const int krow = sLut[kg >> 6] * 64 + (kg & 63);
      v2f a, bb;
      a.x = sS[(m0 + fm) * 385 + kg];
      a.y = sS[(m0 + fm) * 385 + kg + 1];
      bb.x = vp[(size_t)krow * D_ + n0 + fm];
      bb.y = vp[(size_t)(krow + 1) * D_ + n0 + fm];
      acc = wmma4(a, bb, acc);
    }
    const int N = lane & 15, mofs = (lane < 16) ? 0 : 8;
    float* op = o + (size_t)bh * L_ * D_ + (size_t)qrow0 * D_;
    for (int r = 0; r < 8; ++r)
      op[(size_t)(m0 + mofs + r) * D_ + n0 + N] = acc[r];
  }
}

// ---------------------------------------------------------------------------
// K5: per key block: c_k = softmax(k) over D; kvb = c_k^T @ v (64x64);
// ksum = column sums of c_k.
// ---------------------------------------------------------------------------
__global__ __launch_bounds__(256) void k_kvblock(
    const float* __restrict__ k, const float* __restrict__ v,
    float* __restrict__ kvb, float* __restrict__ ksum) {
  const int g = blockIdx.x;                  // g = bh*64 + m
  const float* kp = k + (size_t)g * 4096;
  const float* vp = v + (size_t)g * 4096;
  __shared__ float sCK[64 * 65];
  const int t = threadIdx.x, lane = t & 31, wave = t >> 5;
  const int fm = lane & 15, fkb = (lane >> 4) << 1;
  if (t < 64) {
    float mx = -3.0e38f;
    for (int d = 0; d < 64; ++d) { float vv = kp[t * 64 + d]; if (vv > mx) mx = vv; }
    float sum = 0.f;
    for (int d = 0; d < 64; ++d) {
      float e = __expf(kp[t * 64 + d] - mx); sCK[t * 65 + d] = e; sum += e;
    }
    float inv = 1.f / sum;
    for (int d = 0; d < 64; ++d) sCK[t * 65 + d] *= inv;
  }
  __syncthreads();
  if (t < 64) {
    float s = 0.f;
    for (int kk = 0; kk < 64; ++kk) s += sCK[kk * 65 + t];
    ksum[(size_t)g * 64 + t] = s;
  }
  for (int s2 = 0; s2 < 2; ++s2) {
    const int sub = wave * 2 + s2;
    const int m0 = (sub >> 2) << 4, n0 = (sub & 3) << 4;
    v8f acc = vzero8();
    for (int kk = 0; kk < 64; kk += 4) {
      v2f a, bb;
      a.x = sCK[(kk + fkb) * 65 + m0 + fm];        // A[d][kk] = c_k[kk][d]
      a.y = sCK[(kk + fkb + 1) * 65 + m0 + fm];
      bb.x = vp[(size_t)(kk + fkb) * 64 + n0 + fm];
      bb.y = vp[(size_t)(kk + fkb + 1) * 64 + n0 + fm];
      acc = wmma4(a, bb, acc);
    }
    const int N = lane & 15, mofs = (lane < 16) ? 0 : 8;
    for (int r = 0; r < 8; ++r)
      kvb[(size_t)g * 4096 + (size_t)(m0 + mofs + r) * 64 + n0 + N] = acc[r];
  }
}

// ---------------------------------------------------------------------------
// K6: totals over all 64 key blocks (complement trick: comp = total - topk).
// ---------------------------------------------------------------------------
__global__ __launch_bounds__(256) void k_totals(
    const float* __restrict__ kvb, const float* __restrict__ ksum,
    float* __restrict__ kvtot, float* __restrict__ kstot) {
  const int bh = blockIdx.x, t = threadIdx.x;
  for (int i = t; i < 4096; i += 256) {
    float s = 0.f;
    for (int m = 0; m < 64; ++m) s += kvb[(size_t)bh * 262144 + (size_t)m * 4096 + i];
    kvtot[(size_t)bh * 4096 + i] = s;
  }
  if (t < 64) {
    float s = 0.f;
    for (int m = 0; m < 64; ++m) s += ksum[(size_t)bh * 4096 + m * 64 + t];
    kstot[(size_t)bh * 64 + t] = s;
  }
}

// ---------------------------------------------------------------------------
// K7: linear path per (b,h,n): kv_q = tot - sum(topk); c_q softmax; WMMA
// num = c_q @ kv_q; den; normalize; WMMA proj_l; accumulate into o (+bias).
// ---------------------------------------------------------------------------
__global__ __launch_bounds__(256) void k_linear(
    const float* __restrict__ q, const float* __restrict__ kvb,
    const float* __restrict__ kvtot, const float* __restrict__ ksum,
    const float* __restrict__ kstot, const int* __restrict__ lut,
    const float* __restrict__ plw, const float* __restrict__ plb,
    float* __restrict__ o) {
  const int g = blockIdx.x;                   // g = bh*64 + n
  const int bh = g >> 6;
  __shared__ float sKV[64 * 65];
  __shared__ float sCQ[64 * 65];
  __shared__ float sO[64 * 65];
  __shared__ float sKS[64];
  __shared__ float sDEN[64];
  __shared__ int   sLut[T_];
  const int t = threadIdx.x, lane = t & 31, wave = t >> 5;
  const int fm = lane & 15, fkb = (lane >> 4) << 1;
  if (t < T_) sLut[t] = lut[(size_t)g * T_ + t];
  __syncthreads();
  for (int i = t; i < 4096; i += 256) {
    float s = kvtot[(size_t)bh * 4096 + i];
    for (int j = 0; j < T_; ++j)
      s -= kvb[(size_t)bh * 262144 + (size_t)sLut[j] * 4096 + i];
    sKV[(i >> 6) * 65 + (i & 63)] = s;
  }
  if (t < 64) {
    float s = kstot[(size_t)bh * 64 + t];
    for (int j = 0; j < T_; ++j) s -= ksum[(size_t)bh * 4096 + sLut[j] * 64 + t];
    sKS[t] = s;
  }
  const float* qp = q + (size_t)g * 4096;
  if (t < 64) {
    float mx = -3.0e38f;
    for (int d = 0; d < 64; ++d) { float vv = qp[t * 64 + d]; if (vv > mx) mx = vv; }
    float sum = 0.f;
    for (int d = 0; d < 64; ++d) {
      float e = __expf(qp[t * 64 + d] - mx); sCQ[t * 65 + d] = e; sum += e;
    }
    float inv = 1.f / sum;
    for (int d = 0; d < 64; ++d) sCQ[t * 65 + d] *= inv;
  }
  __syncthreads();
  if (t < 64) {
    float s = 0.f;
    for (int d = 0; d < 64; ++d) s += sCQ[t * 65 + d] * sKS[d];
    sDEN[t] = s;
  }
  for (int s2 = 0; s2 < 2; ++s2) {       // num = c_q @ kv_q
    const int sub = wave * 2 + s2;
    const int m0 = (sub >> 2) << 4, n0 = (sub & 3) << 4;
    v8f acc = vzero8();
    for (int kk = 0; kk < 64; kk += 4) {
      v2f a, bb;
      a.x = sCQ[(m0 + fm) * 65 + kk + fkb];
      a.y = sCQ[(m0 + fm) * 65 + kk + fkb + 1];
      bb.x = sKV[(kk + fkb) * 65 + n0 + fm];
      bb.y = sKV[(kk + fkb + 1) * 65 + n0 + fm];
      acc = wmma4(a, bb, acc);
    }
    const int N = lane & 15, mofs = (lane < 16) ? 0 : 8;
    for (int r = 0; r < 8; ++r) sO[(m0 + mofs + r) * 65 + n0 + N] = acc[r];
  }
  __syncthreads();
  for (int i = t; i < 4096; i += 256) {
    int rr = i >> 6, e = i & 63;
    sO[rr * 65 + e] = sO[rr * 65 + e] / (sDEN[rr] + 1e-6f);
  }
  __syncthreads();
  float* op = o + (size_t)g * 4096;
  for (int s2 = 0; s2 < 2; ++s2) {       // o_l @ proj_lw^T + bias, add into o
    const int sub = wave * 2 + s2;
    const int m0 = (sub >> 2) << 4, n0 = (sub & 3) << 4;
    v8f acc = vzero8();
    for (int kk = 0; kk < 64; kk += 4) {
      v2f a, bb;
      a.x = sO[(m0 + fm) * 65 + kk + fkb];
      a.y = sO[(m0 + fm) * 65 + kk + fkb + 1];
      bb.x = plw[(size_t)(n0 + fm) * 64 + kk + fkb];     // B[d][e] = plw[e][d]
      bb.y = plw[(size_t)(n0 + fm) * 64 + kk + fkb + 1];
      acc = wmma4(a, bb, acc);
    }
    const int N = lane & 15, mofs = (lane < 16) ? 0 : 8;
    for (int r = 0; r < 8; ++r) {
      int row = m0 + mofs + r, e = n0 + N;
      op[(size_t)row * 64 + e] += acc[r] + plb[e];
    }
  }
}

// ---------------------------------------------------------------------------
// K8: out = out_w(256x256) @ o(b, 256, 4096); o channel c = h*64 + d.
// ---------------------------------------------------------------------------
__global__ __launch_bounds__(256) void k_outproj(
    const float* __restrict__ o, const float* __restrict__ w,
    float* __restrict__ out) {
  const int b = blockIdx.z, cy = blockIdx.y, lx = blockIdx.x;
  const int c0 = cy * 64, l0 = lx * 64;
  __shared__ float lW[64 * 65];
  __shared__ float lO[64 * 65];
  const int t = threadIdx.x, lane = t & 31, wave = t >> 5;
  const int fm = lane & 15, fkb = (lane >> 4) << 1;
  v8f acc[2]; acc[0] = vzero8(); acc[1] = vzero8();
  for (int kc = 0; kc < 4; ++kc) {        // K-chunk = head kc (c2 = kc*64 + d)
    __syncthreads();
    for (int r = 0; r < 16; ++r) {
      int idx = r * 256 + t;
      int i = idx >> 6, j = idx & 63;
      lW[i * 65 + j] = w[(size_t)(c0 + i) * C_ + kc * 64 + j];
      lO[j * 65 + i] = o[(((size_t)b * NH_ + kc) * L_ + l0 + i) * D_ + j];
      if (kc < 3)
        __builtin_prefetch(&o[(((size_t)b * NH_ + kc + 1) * L_ + l0 + i) * D_ + j], 0, 0);
    }
    __syncthreads();
    for (int kk = 0; kk < 64; kk += 4) {
      for (int s = 0; s < 2; ++s) {
        int sub = wave * 2 + s;
        int m0 = (sub >> 2) << 4, n0 = (sub & 3) << 4;
        v2f a, bb;
        a.x = lW[(m0 + fm) * 65 + kk + fkb];
        a.y = lW[(m0 + fm) * 65 + kk + fkb + 1];
        bb.x = lO[(kk + fkb) * 65 + n0 + fm];
        bb.y = lO[(kk + fkb + 1) * 65 + n0 + fm];
        acc[s] = wmma4(a, bb, acc[s]);
      }
    }
  }
  const int N = lane & 15, mofs = (lane < 16) ? 0 : 8;
  for (int s = 0; s < 2; ++s) {
    int sub = wave * 2 + s;
    int m0 = (sub >> 2) << 4, n0 = (sub & 3) << 4;
    for (int r = 0; r < 8; ++r)
      out[((size_t)b * C_ + c0 + m0 + mofs + r) * L_ + l0 + n0 + N] = acc[s][r];
  }
}

// ---------------------------------------------------------------------------
extern "C" void kernel_launch(void* const* d_in, const int* in_sizes, int n_in,
                              void* d_out, int out_size, void* d_ws, size_t ws_size,
                              hipStream_t stream) {
  const float* x    = (const float*)d_in[0];  // (8,256,64,64)
  const float* qkvw = (const float*)d_in[1];  // (768,256)
  const float* outw = (const float*)d_in[2];  // (256,256)
  const float* plw  = (const float*)d_in[3];  // (64,64)
  const float* plb  = (const float*)d_in[4];  // (64,)
  float* out = (float*)d_out;                 // (8,256,64,64)

  const size_t QS = (size_t)B_ * NH_ * L_ * D_;     // 8,388,608 floats
  float* ws    = (float*)d_ws;                      // needs ~170 MB
  float* q_    = ws;
  float* k_    = ws + QS;
  float* v_    = ws + 2 * QS;
  float* o_    = ws + 3 * QS;
  float* kvb_  = ws + 4 * QS;                       // (B,NH,nK,D,D) == QS
  float* kvtot_ = ws + 5 * QS;                      // (B,NH,D,D)  131072
  float* ksum_  = kvtot_ + (size_t)B_ * NH_ * D_ * D_;     // (B,NH,nK,D)
  float* kstot_ = ksum_ + (size_t)B_ * NH_ * NK_ * D_;     // (B,NH,D)
  float* qbm_   = kstot_ + (size_t)B_ * NH_ * D_;          // (B,NH,nQ,D)
  float* kbm_   = qbm_ + (size_t)B_ * NH_ * NQ_ * D_;      // (B,NH,nK,D)
  int*   lut_   = (int*)(kbm_ + (size_t)B_ * NH_ * NK_ * D_);  // (B,NH,nQ,6)

  k_qkv    <<<dim3(64, 12, 8),        256, 0, stream>>>(x, qkvw, q_, k_, v_);
  k_bmean  <<<dim3(B_ * NH_ * NQ_),    64, 0, stream>>>(q_, k_, qbm_, kbm_);
  k_topk   <<<dim3(B_ * NH_),          64, 0, stream>>>(qbm_, kbm_, lut_);
  k_sattn  <<<dim3(B_ * NH_ * NQ_ * 2),256, 0, stream>>>(q_, k_, v_, lut_, o_);
  k_kvblock<<<dim3(B_ * NH_ * NK_),   256, 0, stream>>>(k_, v_, kvb_, ksum_);
  k_totals <<<dim3(B_ * NH_),         256, 0, stream>>>(kvb_, ksum_, kvtot_, kstot_);
  k_linear <<<dim3(B_ * NH_ * NQ_),   256, 0, stream>>>(q_, kvb_, kvtot_, ksum_,
                                                        kstot_, lut_, plw, plb, o_);
  k_outproj<<<dim3(64, 4, 8),         256, 0, stream>>>(o_, outw, out);
}